// SumOfKlLoss_3616362463298
// MI455X (gfx1250) — compile-verified
//
#include <hip/hip_runtime.h>

typedef float v2f __attribute__((ext_vector_type(2)));
typedef float v8f __attribute__((ext_vector_type(8)));

#ifndef LN2F
#define LN2F 0.69314718055994530942f
#endif

// KL term per element:
//   vr = (s1/s2)^2 ; t1 = ((m1-m2)/s2)^2
//   kl = 0.5*(vr + t1 - 1 - log(vr)) = 0.5*(vr + t1 - 1) - log(s1/s2)
__device__ __forceinline__ float kl_elem(float m1, float s1, float m2, float s2) {
    float rs2   = __builtin_amdgcn_rcpf(s2);        // v_rcp_f32 (s2 in [0.5,1.5], safe)
    float ratio = s1 * rs2;
    float diff  = (m1 - m2) * rs2;
    float lg    = __builtin_amdgcn_logf(ratio) * LN2F;  // v_log_f32 (log2) -> ln
    return 0.5f * (ratio * ratio + diff * diff - 1.0f) - lg;
}

__global__ void kl_zero_kernel(float* __restrict__ out, int n) {
    int i = blockIdx.x * blockDim.x + threadIdx.x;
    if (i < n) out[i] = 0.0f;
}

// Block: 256 threads = 8 waves. Each wave owns 16 output columns.
// Each wave consumes 4 rows per WMMA step: lane supplies B-slots
//   b[0] -> row (r + 2*(lane>>4) + 0), col = colbase + (lane&15)
//   b[1] -> row (r + 2*(lane>>4) + 1), col = colbase + (lane&15)
// A = ones(16x4)  =>  every row of D accumulates the column sums of B.
__global__ __launch_bounds__(256) void kl_sum_wmma_kernel(
    const float* __restrict__ m1, const float* __restrict__ s1,
    const float* __restrict__ m2, const float* __restrict__ s2,
    float* __restrict__ out, int D, int rows_per_block)
{
    const int lane  = threadIdx.x & 31;
    const int wave  = threadIdx.x >> 5;
    const int col   = blockIdx.x * 128 + wave * 16 + (lane & 15);
    const int khalf = (lane >> 4) << 1;              // 0 or 2
    const int r0    = blockIdx.y * rows_per_block;

    v2f a; a[0] = 1.0f; a[1] = 1.0f;                 // A = ones (16x4)
    v8f c = {};                                      // f32 accumulator (16x16)

    const int rend = r0 + rows_per_block;
    #pragma unroll 4
    for (int r = r0; r < rend; r += 4) {
        const int i0 = (r + khalf) * D + col;        // row r+khalf
        const int i1 = i0 + D;                       // row r+khalf+1

        // streamed once through 192MB L2 -> non-temporal loads
        float a0 = __builtin_nontemporal_load(m1 + i0);
        float b0 = __builtin_nontemporal_load(s1 + i0);
        float c0 = __builtin_nontemporal_load(m2 + i0);
        float d0 = __builtin_nontemporal_load(s2 + i0);
        float a1 = __builtin_nontemporal_load(m1 + i1);
        float b1 = __builtin_nontemporal_load(s1 + i1);
        float c1 = __builtin_nontemporal_load(m2 + i1);
        float d1 = __builtin_nontemporal_load(s2 + i1);

        v2f b;
        b[0] = kl_elem(a0, b0, c0, d0);
        b[1] = kl_elem(a1, b1, c1, d1);

        // D = A(16x4,ones) x B(4x16,kl) + C  -> column sums accumulate
        c = __builtin_amdgcn_wmma_f32_16x16x4_f32(
                /*neg_a=*/false, a, /*neg_b=*/false, b,
                /*c_mod=*/(short)0, c, /*reuse_a=*/false, /*reuse_b=*/false);
    }

    // C/D layout: VGPR0, lanes 0..15 hold (M=0, N=lane) = column sums.
    if (lane < 16) {
        atomicAdd(&out[col], c[0]);
    }
}

extern "C" void kernel_launch(void* const* d_in, const int* in_sizes, int n_in,
                              void* d_out, int out_size, void* d_ws, size_t ws_size,
                              hipStream_t stream) {
    const float* mean1 = (const float*)d_in[0];
    const float* std1  = (const float*)d_in[1];
    const float* mean2 = (const float*)d_in[2];
    const float* std2  = (const float*)d_in[3];
    float* out = (float*)d_out;

    const int D = out_size;               // 2048
    const int B = in_sizes[0] / D;        // 16384

    // zero the (poisoned) output accumulator
    kl_zero_kernel<<<(D + 255) / 256, 256, 0, stream>>>(out, D);

    // 16 column-blocks (128 cols each) x 128 row-blocks (128 rows each)
    const int ROW_SPLIT = 128;
    const int rows_per_block = B / ROW_SPLIT;   // 128
    dim3 grid(D / 128, ROW_SPLIT);
    kl_sum_wmma_kernel<<<grid, 256, 0, stream>>>(mean1, std1, mean2, std2,
                                                 out, D, rows_per_block);
}